// SpinSphericalConvolution_55473797595153
// MI455X (gfx1250) — compile-verified
//
#include <hip/hip_runtime.h>

// Spin-weighted spherical convolution for MI455X (gfx1250), fp32 WMMA pipeline.
// Stages: DFT(phi) -> forward Legendre -> complex channel mix -> backward
// Legendre -> inverse DFT(phi), all as batched 16x16-tile WMMA matmuls using
// V_WMMA_F32_16X16X4_F32 (full fp32: problem is HBM-bound, ~268MB / ~30GFLOP,
// so low-precision WMMA buys nothing and would hurt vs complex64 reference).
//
// Round-2 refinements:
//  * register-block 4 (or 2) output tiles per wave -> A-fragments and their
//    negations loaded/computed once per K-step, reused across tiles
//  * stage-3 B matrix stored K-contiguous (transposed) -> b64 B-frag loads
//  * stages 1/5 stage their 64x64 complex tile in LDS *transposed* with pad 66
//    (8B-aligned pairs -> ds_load_b64, even-bank spread, no conflicts)
//  * stage-5 synthesizes the zero m=+/-32 pad row in LDS (zero-pad kernel gone)

typedef __attribute__((ext_vector_type(2))) float v2f;
typedef __attribute__((ext_vector_type(8))) float v8f;

#define LDS_PAD 66  // even pad: keeps K-pairs 8B aligned AND banks spread

__device__ __forceinline__ v8f wmma4(v2f a, v2f b, v8f c) {
  // D = A(16x4) * B(4x16) + C, fp32
  return __builtin_amdgcn_wmma_f32_16x16x4_f32(false, a, false, b, (short)0, c,
                                               false, false);
}

// A fragment (16x4): lane = M row; lanes 0-15 hold K={k0,k0+1}, 16-31 K={k0+2,k0+3}
// Also used for K-contiguous ("transposed") B storage: element (k, col) at
// col*ld + k, which is the identical access pattern.
template <typename PT>
__device__ __forceinline__ v2f frag_a(PT A, int lda, int k0) {
  int lane = threadIdx.x & 31;
  int row = lane & 15;
  int kk = k0 + ((lane & 16) >> 3);
  v2f r;
  r.x = A[row * lda + kk];
  r.y = A[row * lda + kk + 1];
  return r;
}

// B fragment from row-major (N-contiguous) storage: two strided b32 loads
template <typename PT>
__device__ __forceinline__ v2f frag_b(PT Bm, int ldb, int k0) {
  int lane = threadIdx.x & 31;
  int col = lane & 15;
  int kk = k0 + ((lane & 16) >> 3);
  v2f r;
  r.x = Bm[kk * ldb + col];
  r.y = Bm[(kk + 1) * ldb + col];
  return r;
}

__device__ __forceinline__ v2f vneg(v2f a) {
  v2f r; r.x = -a.x; r.y = -a.y; return r;
}

// C/D tile (16x16 f32): VGPR i -> row i (+8 for upper half-wave), lane = N col
__device__ __forceinline__ void store_tile(float* C, int ldc, v8f acc,
                                           int rows_valid) {
  int lane = threadIdx.x & 31;
  int col = lane & 15;
  int rb = (lane & 16) >> 1;  // 0 or 8
#pragma unroll
  for (int i = 0; i < 8; ++i) {
    int row = rb + i;
    if (row < rows_valid) C[(size_t)row * ldc + col] = acc[i];
  }
}

// ---- table builders --------------------------------------------------------

// D[m][p] = exp(-2pi*i*(m-31)*p/64)/64 (row 63 = 0); E[p][m] = exp(+...)
__global__ void k_build_dft(float* Dre, float* Dim, float* Ere, float* Eim) {
  int idx = blockIdx.x * blockDim.x + threadIdx.x;
  if (idx >= 4096) return;
  int r = idx >> 6, c = idx & 63;
  const float TP = 6.283185307179586f;
  {
    float vre = 0.f, vim = 0.f;
    if (r < 63) {
      float ang = -TP * (float)(r - 31) * (float)c * (1.0f / 64.0f);
      vre = cosf(ang) * (1.0f / 64.0f);
      vim = sinf(ang) * (1.0f / 64.0f);
    }
    Dre[idx] = vre; Dim[idx] = vim;
  }
  {
    float vre = 0.f, vim = 0.f;
    if (c < 63) {
      float ang = TP * (float)(c - 31) * (float)r * (1.0f / 64.0f);
      vre = cosf(ang);
      vim = sinf(ang);
    }
    Ere[idx] = vre; Eim[idx] = vim;
  }
}

// WLin[((s*63+m)*32+l)*64+j] = 2pi*w[j]*leg_in[s,l,m,j]
// LOt [((s*63+m)*64+j)*32+l] = leg_out[s,l,m,j]
__global__ void k_build_wleg(const float* __restrict__ leg_in,
                             const float* __restrict__ leg_out,
                             const float* __restrict__ w,
                             float* __restrict__ WLin, float* __restrict__ LOt) {
  int idx = blockIdx.x * blockDim.x + threadIdx.x;
  if (idx >= 258048) return;
  {
    int j = idx & 63, l = (idx >> 6) & 31;
    int rest = idx >> 11, m = rest % 63, s = rest / 63;
    WLin[idx] = 6.283185307179586f * w[j] *
                leg_in[(((size_t)s * 32 + l) * 63 + m) * 64 + j];
  }
  {
    int l = idx & 31, j = (idx >> 5) & 63;
    int rest = idx >> 11, m = rest % 63, s = rest / 63;
    LOt[idx] = leg_out[(((size_t)s * 32 + l) * 63 + m) * 64 + j];
  }
}

// KmatT[l][n=(o*64+d)][k=(i*64+c)] = kernel[l,i,o,c,d]  (K-contiguous!)
__global__ void k_build_kmatT(const float* __restrict__ kre,
                              const float* __restrict__ kim,
                              float* __restrict__ Kr, float* __restrict__ Ki) {
  int idx = blockIdx.x * blockDim.x + threadIdx.x;
  if (idx >= 524288) return;
  int l = idx >> 14, n = (idx >> 7) & 127, k = idx & 127;
  int i = k >> 6, c = k & 63, o = n >> 6, d = n & 63;
  size_t src = ((((size_t)l * 2 + i) * 2 + o) * 64 + c) * 64 + d;
  Kr[idx] = kre[src];
  Ki[idx] = kim[src];
}

// ---- stage 1: forward DFT over phi  g[m,c] = D[m,p] * F[p,c] (complex) -----
__global__ __launch_bounds__(128) void k_stage1_dft(
    const float* __restrict__ f_re, const float* __restrict__ f_im,
    const float* __restrict__ Dre, const float* __restrict__ Dim,
    float* __restrict__ g_re, float* __restrict__ g_im) {
  __shared__ float Tre[64 * LDS_PAD];  // transposed: Tre[c][p]
  __shared__ float Tim[64 * LDS_PAD];
  int slice = blockIdx.x;  // ((b*64+j)*2+s)
  int s = slice & 1;
  int bj = slice >> 1;
  int j = bj & 63, b = bj >> 6;
  size_t base = (size_t)bj * 8192 + (size_t)s * 64;  // f[b][j][p][s][c]
  for (int e = threadIdx.x; e < 4096; e += 128) {
    int p = e >> 6, c = e & 63;  // coalesced global read, transposed LDS write
    Tre[c * LDS_PAD + p] = f_re[base + (size_t)p * 128 + c];
    Tim[c * LDS_PAD + p] = f_im[base + (size_t)p * 128 + c];
  }
  __syncthreads();
  size_t gb = ((size_t)(b * 2 + s) * 63) * 4096 + (size_t)j * 64;
  int mt = threadIdx.x >> 5;  // wave owns one row tile, 4 col tiles
  const float* Ar = Dre + mt * 16 * 64;
  const float* Ai = Dim + mt * 16 * 64;
  v8f cr[4] = {}; v8f ci[4] = {};
#pragma unroll
  for (int kk = 0; kk < 64; kk += 4) {
    v2f ar = frag_a(Ar, 64, kk);
    v2f ai = frag_a(Ai, 64, kk);
    v2f nai = vneg(ai);
#pragma unroll
    for (int ct = 0; ct < 4; ++ct) {
      v2f br = frag_a(&Tre[ct * 16 * LDS_PAD], LDS_PAD, kk);  // ds_load_b64
      v2f bi = frag_a(&Tim[ct * 16 * LDS_PAD], LDS_PAD, kk);
      cr[ct] = wmma4(ar, br, cr[ct]);
      cr[ct] = wmma4(nai, bi, cr[ct]);
      ci[ct] = wmma4(ar, bi, ci[ct]);
      ci[ct] = wmma4(ai, br, ci[ct]);
    }
  }
  int rows_valid = 63 - mt * 16;  // m rows 0..62 only
#pragma unroll
  for (int ct = 0; ct < 4; ++ct) {
    store_tile(g_re + gb + (size_t)mt * 16 * 4096 + ct * 16, 4096, cr[ct], rows_valid);
    store_tile(g_im + gb + (size_t)mt * 16 * 4096 + ct * 16, 4096, ci[ct], rows_valid);
  }
}

// ---- stage 2: forward Legendre  A[l,c] = WL[l,j] * g[j,c] (real x complex) -
__global__ __launch_bounds__(128) void k_stage2_leg(
    const float* __restrict__ WLin, const float* __restrict__ g_re,
    const float* __restrict__ g_im, float* __restrict__ a_re,
    float* __restrict__ a_im) {
  int slice = blockIdx.x;  // ((b*63+m)*2+s)
  int s = slice & 1;
  int bm = slice >> 1;
  int m = bm % 63, b = bm / 63;
  size_t gb = ((size_t)(b * 2 + s) * 63 + m) * 4096;    // g[j][c], ldb=64
  size_t ab = (size_t)b * 258048 + (size_t)m * 128 + (size_t)s * 64;  // l stride 8064
  int wave = threadIdx.x >> 5;
  int lt = wave >> 1;           // 0..1
  int cg = (wave & 1) * 2;      // col-tile group base: 0 or 2
  const float* At = WLin + (size_t)(s * 63 + m) * 2048 + lt * 16 * 64;
  v8f cr[2] = {}; v8f ci[2] = {};
#pragma unroll
  for (int kk = 0; kk < 64; kk += 4) {
    v2f a = frag_a(At, 64, kk);
#pragma unroll
    for (int q = 0; q < 2; ++q) {
      v2f br = frag_b(g_re + gb + (cg + q) * 16, 64, kk);
      v2f bi = frag_b(g_im + gb + (cg + q) * 16, 64, kk);
      cr[q] = wmma4(a, br, cr[q]);
      ci[q] = wmma4(a, bi, ci[q]);
    }
  }
#pragma unroll
  for (int q = 0; q < 2; ++q) {
    store_tile(a_re + ab + (size_t)lt * 16 * 8064 + (cg + q) * 16, 8064, cr[q], 16);
    store_tile(a_im + ab + (size_t)lt * 16 * 8064 + (cg + q) * 16, 8064, ci[q], 16);
  }
}

// ---- stage 3: channel/spin mix  Aout[b,n] = Ain[b,k] * K_l[k,n] (complex) --
__global__ __launch_bounds__(128) void k_stage3_mix(
    const float* __restrict__ ain_re, const float* __restrict__ ain_im,
    const float* __restrict__ Ktr, const float* __restrict__ Kti,
    float* __restrict__ aout_re, float* __restrict__ aout_im) {
  int slice = blockIdx.x;  // l*63+m
  size_t abase = (size_t)slice * 128;  // + b*258048 + k
  const float* Btr = Ktr + (size_t)(slice / 63) * 16384;  // [n][k] K-contig
  const float* Bti = Kti + (size_t)(slice / 63) * 16384;
  int wave = threadIdx.x >> 5;
  int bt = wave >> 1;           // 0..1
  int ng = (wave & 1) * 4;      // n-tile group base: 0 or 4
  const float* Ar = ain_re + abase + (size_t)bt * 16 * 258048;
  const float* Ai = ain_im + abase + (size_t)bt * 16 * 258048;
  v8f cr[4] = {}; v8f ci[4] = {};
  for (int kk = 0; kk < 128; kk += 4) {
    v2f ar = frag_a(Ar, 258048, kk);
    v2f ai = frag_a(Ai, 258048, kk);
    v2f nai = vneg(ai);
#pragma unroll
    for (int q = 0; q < 4; ++q) {
      v2f br = frag_a(Btr + (ng + q) * 16 * 128, 128, kk);  // b64 loads
      v2f bi = frag_a(Bti + (ng + q) * 16 * 128, 128, kk);
      cr[q] = wmma4(ar, br, cr[q]);
      cr[q] = wmma4(nai, bi, cr[q]);
      ci[q] = wmma4(ar, bi, ci[q]);
      ci[q] = wmma4(ai, br, ci[q]);
    }
  }
#pragma unroll
  for (int q = 0; q < 4; ++q) {
    store_tile(aout_re + abase + (size_t)bt * 16 * 258048 + (ng + q) * 16,
               258048, cr[q], 16);
    store_tile(aout_im + abase + (size_t)bt * 16 * 258048 + (ng + q) * 16,
               258048, ci[q], 16);
  }
}

// ---- stage 4: backward Legendre  H[j,d] = LO[j,l] * Aout[l,d] --------------
__global__ __launch_bounds__(128) void k_stage4_legT(
    const float* __restrict__ LOt, const float* __restrict__ aout_re,
    const float* __restrict__ aout_im, float* __restrict__ h_re,
    float* __restrict__ h_im) {
  int slice = blockIdx.x;  // ((b*63+m)*2+s)
  int s = slice & 1;
  int bm = slice >> 1;
  int m = bm % 63, b = bm / 63;
  size_t bb = (size_t)b * 258048 + (size_t)m * 128 + (size_t)s * 64;  // l stride 8064
  size_t hb = (size_t)(b * 2 + s) * 262144 + (size_t)m * 64;          // j stride 4096
  int jt = threadIdx.x >> 5;  // wave owns one j tile, 4 d tiles
  const float* At = LOt + (size_t)(s * 63 + m) * 2048 + jt * 16 * 32;  // 64x32
  v8f cr[4] = {}; v8f ci[4] = {};
#pragma unroll
  for (int kk = 0; kk < 32; kk += 4) {
    v2f a = frag_a(At, 32, kk);
#pragma unroll
    for (int dt = 0; dt < 4; ++dt) {
      v2f br = frag_b(aout_re + bb + dt * 16, 8064, kk);
      v2f bi = frag_b(aout_im + bb + dt * 16, 8064, kk);
      cr[dt] = wmma4(a, br, cr[dt]);
      ci[dt] = wmma4(a, bi, ci[dt]);
    }
  }
#pragma unroll
  for (int dt = 0; dt < 4; ++dt) {
    store_tile(h_re + hb + (size_t)jt * 16 * 4096 + dt * 16, 4096, cr[dt], 16);
    store_tile(h_im + hb + (size_t)jt * 16 * 4096 + dt * 16, 4096, ci[dt], 16);
  }
}

// ---- stage 5: inverse DFT  out[p,d] = E[p,m] * H[m,d], interleaved store ---
__global__ __launch_bounds__(128) void k_stage5_idft(
    const float* __restrict__ Ere, const float* __restrict__ Eim,
    const float* __restrict__ h_re, const float* __restrict__ h_im,
    float* __restrict__ out) {
  __shared__ float Tre[64 * LDS_PAD];  // transposed: Tre[d][m], m=63 row zeroed
  __shared__ float Tim[64 * LDS_PAD];
  int slice = blockIdx.x;  // ((b*64+j)*2+s)
  int s = slice & 1;
  int bj = slice >> 1;
  int j = bj & 63, b = bj >> 6;
  size_t hb = (size_t)(b * 2 + s) * 262144 + (size_t)j * 4096;  // h[m][d], ld=64
  for (int e = threadIdx.x; e < 4096; e += 128) {
    int mm = e >> 6, d = e & 63;  // coalesced read; pad row m=63 -> 0 in LDS
    float vr = 0.f, vi = 0.f;
    if (mm < 63) {
      vr = h_re[hb + (size_t)mm * 64 + d];
      vi = h_im[hb + (size_t)mm * 64 + d];
    }
    Tre[d * LDS_PAD + mm] = vr;
    Tim[d * LDS_PAD + mm] = vi;
  }
  __syncthreads();
  size_t ob = (size_t)bj * 8192 + (size_t)s * 64;  // complex elem base
  int pt = threadIdx.x >> 5;  // wave owns one p tile, 4 d tiles
  const float* Ar = Ere + pt * 16 * 64;
  const float* Ai = Eim + pt * 16 * 64;
  v8f cr[4] = {}; v8f ci[4] = {};
#pragma unroll
  for (int kk = 0; kk < 64; kk += 4) {
    v2f ar = frag_a(Ar, 64, kk);
    v2f ai = frag_a(Ai, 64, kk);
    v2f nai = vneg(ai);
#pragma unroll
    for (int dt = 0; dt < 4; ++dt) {
      v2f br = frag_a(&Tre[dt * 16 * LDS_PAD], LDS_PAD, kk);  // ds_load_b64
      v2f bi = frag_a(&Tim[dt * 16 * LDS_PAD], LDS_PAD, kk);
      cr[dt] = wmma4(ar, br, cr[dt]);
      cr[dt] = wmma4(nai, bi, cr[dt]);
      ci[dt] = wmma4(ar, bi, ci[dt]);
      ci[dt] = wmma4(ai, br, ci[dt]);
    }
  }
  int lane = threadIdx.x & 31;
  int col = lane & 15;
  int rb = (lane & 16) >> 1;
#pragma unroll
  for (int dt = 0; dt < 4; ++dt) {
    float* o = out + 2 * (ob + (size_t)(pt * 16) * 128 + dt * 16);
#pragma unroll
    for (int i = 0; i < 8; ++i) {
      int row = rb + i;
      float2 v = make_float2(cr[dt][i], ci[dt][i]);
      *reinterpret_cast<float2*>(o + 2 * ((size_t)row * 128 + col)) = v;
    }
  }
}

// ---- host ------------------------------------------------------------------
extern "C" void kernel_launch(void* const* d_in, const int* in_sizes, int n_in,
                              void* d_out, int out_size, void* d_ws,
                              size_t ws_size, hipStream_t stream) {
  const float* f_re = (const float*)d_in[0];
  const float* f_im = (const float*)d_in[1];
  const float* k_re = (const float*)d_in[2];
  const float* k_im = (const float*)d_in[3];
  const float* leg_in = (const float*)d_in[4];
  const float* leg_out = (const float*)d_in[5];
  const float* quad_w = (const float*)d_in[6];
  float* ws = (float*)d_ws;

  // workspace layout (floats); h aliases the g/a_in region (dead by stage 4)
  float* g_re = ws + 0;
  float* g_im = ws + 16515072u;
  float* ain_re = ws + 33030144u;
  float* ain_im = ws + 41287680u;
  float* h_re = ws + 0;          // alias
  float* h_im = ws + 16777216u;  // alias
  float* aout_re = ws + 49545216u;
  float* aout_im = ws + 57802752u;
  float* KmatT_re = ws + 66060288u;
  float* KmatT_im = ws + 66584576u;
  float* WLin = ws + 67108864u;
  float* LOt = ws + 67366912u;
  float* Dre = ws + 67624960u;
  float* Dim = ws + 67629056u;
  float* Ere = ws + 67633152u;
  float* Eim = ws + 67637248u;
  // total required: 67641344 floats (~258 MiB) of d_ws

  k_build_dft<<<16, 256, 0, stream>>>(Dre, Dim, Ere, Eim);
  k_build_wleg<<<1008, 256, 0, stream>>>(leg_in, leg_out, quad_w, WLin, LOt);
  k_build_kmatT<<<2048, 256, 0, stream>>>(k_re, k_im, KmatT_re, KmatT_im);

  k_stage1_dft<<<4096, 128, 0, stream>>>(f_re, f_im, Dre, Dim, g_re, g_im);
  k_stage2_leg<<<4032, 128, 0, stream>>>(WLin, g_re, g_im, ain_re, ain_im);
  k_stage3_mix<<<2016, 128, 0, stream>>>(ain_re, ain_im, KmatT_re, KmatT_im,
                                         aout_re, aout_im);
  k_stage4_legT<<<4032, 128, 0, stream>>>(LOt, aout_re, aout_im, h_re, h_im);
  k_stage5_idft<<<4096, 128, 0, stream>>>(Ere, Eim, h_re, h_im, (float*)d_out);
}